// Vid2Seq_77979426226339
// MI455X (gfx1250) — compile-verified
//
#include <hip/hip_runtime.h>
#include <hip/hip_bf16.h>
#include <math.h>

#define B_     32
#define T_     2048
#define D_     768
#define TOPM   8
#define MPAD   16      // ww rows padded to WMMA M=16 (rows 8..15 are zeros)
#define MINLEN 4
#define MAXGAP 2
#define MAXSEG 512

typedef float v2f __attribute__((ext_vector_type(2)));
typedef float v8f __attribute__((ext_vector_type(8)));

// ---------------------------------------------------------------------------
// Kernel 1: per-batch segmentation. One 256-thread workgroup per batch.
// Produces dense softmax attention weights ww[B][MPAD][T] in workspace
// (rows 8..15 zero-padded for WMMA), with fallback (uniform 1/T in row 0)
// and invalid-slot zeroing folded in, so the pooling step is a pure GEMM.
// ---------------------------------------------------------------------------
__global__ __launch_bounds__(256) void vid2seq_weights(const float* __restrict__ w_bt,
                                                       float* __restrict__ ww) {
  __shared__ float w[T_];          // smoothed saliency
  __shared__ float srt[T_];        // sort buffer for quantile
  __shared__ int   msk[T_];
  __shared__ int   segS[MAXSEG], segE[MAXSEG];
  __shared__ float segSc[MAXSEG];
  __shared__ int   nsegS;
  __shared__ int   selS[TOPM], selE[TOPM], selV[TOPM];
  __shared__ float rowMax[TOPM], rowSum[TOPM];
  __shared__ float tauS;

  const int b   = blockIdx.x;
  const int tid = threadIdx.x;
  const float* x = w_bt + b * T_;

  // 1. box filter k=5 with replicate padding
  for (int t = tid; t < T_; t += 256) {
    float s = 0.f;
#pragma unroll
    for (int j = -2; j <= 2; ++j) {
      int u = t + j;
      u = u < 0 ? 0 : (u > T_ - 1 ? T_ - 1 : u);
      s += x[u];
    }
    float v = s * 0.2f;
    w[t] = v;
    srt[t] = v;
  }
  __syncthreads();

  // 2. bitonic sort (ascending) of 2048 elements with 256 threads
  for (int k = 2; k <= T_; k <<= 1) {
    for (int j = k >> 1; j > 0; j >>= 1) {
      for (int i = tid; i < T_; i += 256) {
        int ixj = i ^ j;
        if (ixj > i) {
          float a = srt[i], c = srt[ixj];
          bool up = ((i & k) == 0);
          if (up ? (a > c) : (a < c)) { srt[i] = c; srt[ixj] = a; }
        }
      }
      __syncthreads();
    }
  }

  // 3. quantile(0.7) with linear interpolation
  if (tid == 0) {
    float q = 0.7f * (float)(T_ - 1);
    int lo   = (int)floorf(q);
    float fr = q - (float)lo;
    tauS = srt[lo] + fr * (srt[lo + 1] - srt[lo]);
  }
  __syncthreads();
  const float tau = tauS;

  // 4. threshold mask
  for (int t = tid; t < T_; t += 256) msk[t] = (w[t] >= tau) ? 1 : 0;
  __syncthreads();

  // 5. serial run-length machinery (thread 0; T=2048 is tiny)
  if (tid == 0) {
    // min-length filter
    int t = 0;
    while (t < T_) {
      if (msk[t]) {
        int s0 = t;
        while (t < T_ && msk[t]) ++t;
        if (t - s0 < MINLEN)
          for (int u = s0; u < t; ++u) msk[u] = 0;
      } else ++t;
    }
    // merge: fill internal zero-runs of length <= MAXGAP
    t = 0;
    while (t < T_) {
      if (!msk[t]) {
        int s0 = t;
        while (t < T_ && !msk[t]) ++t;
        if (s0 > 0 && t < T_ && (t - s0) <= MAXGAP)
          for (int u = s0; u < t; ++u) msk[u] = 1;
      } else ++t;
    }
    // collect segments and scores (sum of smoothed w)
    int ns = 0;
    t = 0;
    while (t < T_) {
      if (msk[t]) {
        int s0 = t;
        float sc = 0.f;
        while (t < T_ && msk[t]) { sc += w[t]; ++t; }
        if (ns < MAXSEG) { segS[ns] = s0; segE[ns] = t; segSc[ns] = sc; ++ns; }
      } else ++t;
    }
    nsegS = ns;
    // top-8: descending, first index wins ties (matches lax.top_k)
    for (int r = 0; r < TOPM; ++r) {
      float best = -3.0e38f;
      int bi = -1;
      for (int i = 0; i < ns; ++i)
        if (segSc[i] > best) { best = segSc[i]; bi = i; }
      if (bi >= 0) {
        selS[r] = segS[bi]; selE[r] = segE[bi]; selV[r] = 1;
        segSc[bi] = -3.4e38f;  // consume
      } else {
        selV[r] = 0;
      }
    }
    // per-row softmax statistics
    for (int r = 0; r < TOPM; ++r) {
      if (selV[r]) {
        float mx = -3.0e38f;
        for (int u = selS[r]; u < selE[r]; ++u) mx = fmaxf(mx, w[u]);
        float sm = 0.f;
        for (int u = selS[r]; u < selE[r]; ++u) sm += __expf(w[u] - mx);
        rowMax[r] = mx;
        rowSum[r] = sm;
      }
    }
  }
  __syncthreads();

  // 6. materialize dense weight matrix, padded to 16 rows
  //    (fallback + invalid rows + padding rows folded in)
  const int anyValid = nsegS > 0;
  float* wwb = ww + (size_t)b * MPAD * T_;
  for (int i = tid; i < MPAD * T_; i += 256) {
    int r = i / T_;
    int t = i - r * T_;
    float v = 0.0f;
    if (r < TOPM) {
      if (!anyValid) {
        v = (r == 0) ? (1.0f / T_) : 0.0f;
      } else if (selV[r] && t >= selS[r] && t < selE[r]) {
        v = __expf(w[t] - rowMax[r]) / rowSum[r];
      }
    }
    wwb[i] = v;
  }
}

// ---------------------------------------------------------------------------
// Kernel 2: pooled = ww @ video via V_WMMA_F32_16X16X4_F32 (full fp32).
// One wave32 per (batch, 16-wide column tile of D). ww is pre-padded to
// M=16 so the A-fragment load is an unconditional aligned b64 — no EXEC
// manipulation in the hot loop. Prefetch is unconditional (speculative,
// silently dropped past the end per ISA).
// A layout (16x4 f32): lanes 0-15 -> M=lane, K0/K1 in v0/v1;
//                      lanes 16-31 -> M=lane-16, K2/K3 in v0/v1.
// B layout mirrored: lanes 0-15 -> N=n0+lane, rows K0/K1; hi lanes rows K2/K3.
// D layout: VGPR r, lanes 0-15 -> M=r (rows 8..15 in hi lanes are padding).
// ---------------------------------------------------------------------------
__global__ __launch_bounds__(32) void vid2seq_pool_wmma(const float* __restrict__ ww,
                                                        const float* __restrict__ video,
                                                        float* __restrict__ out) {
  const int NT   = D_ / 16;  // 48 column tiles
  const int b    = blockIdx.x / NT;
  const int nt   = blockIdx.x % NT;
  const int n0   = nt * 16;
  const int lane = threadIdx.x;
  const int mrow = lane & 15;
  const int hi   = lane >> 4;  // 0 -> K{0,1}, 1 -> K{2,3}

  // A pointer: row mrow of padded ww, starting at K = 2*hi; 8-byte aligned.
  const float* aPtr = ww + (size_t)b * MPAD * T_ + (size_t)mrow * T_ + (hi << 1);
  // B pointer: video[k = 2*hi][n0 + mrow] within this batch.
  const float* bPtr = video + (size_t)b * T_ * D_ + (size_t)(hi << 1) * D_ + n0 + mrow;

  v8f acc = {};
#pragma unroll 4
  for (int k0 = 0; k0 < T_; k0 += 4) {
    v2f a = *(const v2f*)aPtr;              // single global_load_b64
    v2f bf;
    bf.x = bPtr[0];                         // video[k][n]
    bf.y = bPtr[D_];                        // video[k+1][n]
    __builtin_prefetch(bPtr + (size_t)64 * D_, 0, 1);  // ~16 iterations ahead
    acc = __builtin_amdgcn_wmma_f32_16x16x4_f32(
        /*neg_a=*/false, a, /*neg_b=*/false, bf,
        /*c_mod=*/(short)0, acc, /*reuse_a=*/false, /*reuse_b=*/false);
    aPtr += 4;
    bPtr += (size_t)4 * D_;
  }

  if (!hi) {  // only M = 0..7 are real output rows
    float* o = out + (size_t)b * TOPM * D_ + n0 + mrow;
#pragma unroll
    for (int r = 0; r < TOPM; ++r) o[(size_t)r * D_] = acc[r];
  }
}

extern "C" void kernel_launch(void* const* d_in, const int* in_sizes, int n_in,
                              void* d_out, int out_size, void* d_ws, size_t ws_size,
                              hipStream_t stream) {
  (void)in_sizes; (void)n_in; (void)out_size; (void)ws_size;
  const float* w_bt  = (const float*)d_in[0];   // [B, T]
  const float* video = (const float*)d_in[1];   // [B, T, D]
  float* out = (float*)d_out;                   // [B, TOPM, D]
  float* ww  = (float*)d_ws;                    // [B, MPAD, T] = 4 MiB scratch

  vid2seq_weights<<<B_, 256, 0, stream>>>(w_bt, ww);
  vid2seq_pool_wmma<<<B_ * (D_ / 16), 32, 0, stream>>>(ww, video, out);
}